// Distance_11759620456454
// MI455X (gfx1250) — compile-verified
//
#include <hip/hip_runtime.h>
#include <math.h>

typedef float v2f __attribute__((ext_vector_type(2)));
typedef float v8f __attribute__((ext_vector_type(8)));

constexpr int kD     = 128;   // feature dim
constexpr int kM     = 512;   // reference rows
constexpr int kR     = 2048;  // query rows (B*N = 4*512)
constexpr int kPad   = 132;   // padded LDS row stride in floats (128+4 -> bank-conflict free)
constexpr int kBT    = 32;    // block tile (32 x-rows x 32 y-rows)

__global__ __launch_bounds__(128) void distance_wmma_kernel(
    const float* __restrict__ X,   // [kR, kD]
    const float* __restrict__ Y,   // [kM, kD]
    float* __restrict__ out)       // [kR, kM, 3] (cos, p1, p2)
{
    __shared__ float sX[kBT * kPad];
    __shared__ float sY[kBT * kPad];
    __shared__ float sXn2[kBT];
    __shared__ float sYn2[kBT];

    const int tid    = threadIdx.x;
    const int wave   = tid >> 5;
    const int lane   = tid & 31;
    const int laneHi = lane >> 4;   // 0/1 : which half of wave
    const int laneLo = lane & 15;   // row/col index 0..15

    const int rbase0 = blockIdx.y * kBT;   // x-row tile origin
    const int mbase0 = blockIdx.x * kBT;   // y-row tile origin

    // ---------------- cooperative global -> LDS (float4) ----------------
    const float4* Xg = (const float4*)(X + (size_t)rbase0 * kD);
    const float4* Yg = (const float4*)(Y + (size_t)mbase0 * kD);
    #pragma unroll
    for (int i = 0; i < 8; ++i) {
        const int slot = tid + i * 128;     // 0..1023 = 32 rows * 32 float4
        const int row  = slot >> 5;
        const int c4   = slot & 31;
        const float4 xv = Xg[row * (kD / 4) + c4];
        const float4 yv = Yg[row * (kD / 4) + c4];
        *(float4*)(&sX[row * kPad + c4 * 4]) = xv;
        *(float4*)(&sY[row * kPad + c4 * 4]) = yv;
    }
    __syncthreads();

    // ---------------- per-row squared norms (waves 0 and 2) ----------------
    if (wave == 0) {
        const float* p = &sX[lane * kPad];
        float s = 0.f;
        #pragma unroll
        for (int j = 0; j < kD / 4; ++j) {
            const float4 t = *(const float4*)(p + j * 4);
            s += t.x * t.x + t.y * t.y + t.z * t.z + t.w * t.w;
        }
        sXn2[lane] = s;
    } else if (wave == 2) {
        const float* p = &sY[lane * kPad];
        float s = 0.f;
        #pragma unroll
        for (int j = 0; j < kD / 4; ++j) {
            const float4 t = *(const float4*)(p + j * 4);
            s += t.x * t.x + t.y * t.y + t.z * t.z + t.w * t.w;
        }
        sYn2[lane] = s;
    }

    // wave -> 16x16 sub-tile within the 32x32 block tile
    const int wr = (wave >> 1) * 16;   // x-row offset
    const int wm = (wave & 1) * 16;    // y-row offset

    // ---------------- dot products via exact f32 WMMA ----------------
    // A fragment (16x4 f32): lane holds row m=laneLo, K = 2*laneHi + vgpr
    // B fragment (4x16 f32): lane holds col n=laneLo, K = 2*laneHi + vgpr
    v8f c = {0.f, 0.f, 0.f, 0.f, 0.f, 0.f, 0.f, 0.f};
    const float* xA = &sX[(wr + laneLo) * kPad + laneHi * 2];
    const float* yB = &sY[(wm + laneLo) * kPad + laneHi * 2];
    #pragma unroll
    for (int kk = 0; kk < kD; kk += 4) {
        const v2f a = *(const v2f*)(xA + kk);
        const v2f b = *(const v2f*)(yB + kk);
        c = __builtin_amdgcn_wmma_f32_16x16x4_f32(
                /*neg_a=*/false, a, /*neg_b=*/false, b,
                /*c_mod=*/(short)0, c, /*reuse_a=*/false, /*reuse_b=*/false);
    }

    // ---------------- L1 distances (VALU streaming from LDS) ----------------
    // C/D layout: vgpr v, lane -> (m = wr + 8*laneHi + v, n = wm + laneLo)
    // The lane's y-row is fixed across all 8 outputs -> hoist chunks to regs.
    float acc[8] = {0.f, 0.f, 0.f, 0.f, 0.f, 0.f, 0.f, 0.f};
    const float* yrow  = &sY[(wm + laneLo) * kPad];
    const float* xrows = &sX[(wr + laneHi * 8) * kPad];
    #pragma unroll
    for (int ch = 0; ch < 4; ++ch) {               // 4 chunks of 32 floats
        float4 yb[8];
        #pragma unroll
        for (int j = 0; j < 8; ++j)
            yb[j] = *(const float4*)(yrow + ch * 32 + j * 4);
        #pragma unroll
        for (int v = 0; v < 8; ++v) {
            const float* xr = xrows + v * kPad + ch * 32;
            float s = 0.f;
            #pragma unroll
            for (int j = 0; j < 8; ++j) {
                const float4 xa = *(const float4*)(xr + j * 4);
                s += fabsf(xa.x - yb[j].x) + fabsf(xa.y - yb[j].y)
                   + fabsf(xa.z - yb[j].z) + fabsf(xa.w - yb[j].w);
            }
            acc[v] += s;
        }
    }

    __syncthreads();   // norms must be visible to all waves

    // ---------------- epilogue: reconstruct cos / p2, store ----------------
    const float yn2 = sYn2[wm + laneLo];
    const int   mg  = mbase0 + wm + laneLo;
    #pragma unroll
    for (int v = 0; v < 8; ++v) {
        const int   ml  = wr + laneHi * 8 + v;
        const float xn2 = sXn2[ml];
        const int   rg  = rbase0 + ml;
        const float dot = c[v];
        const float cosv = dot * rsqrtf(xn2 * yn2);
        const float p2   = sqrtf(fmaxf(xn2 + yn2 - 2.f * dot, 0.f));
        const int   off  = (rg * kM + mg) * 3;
        out[off + 0] = cosv;
        out[off + 1] = acc[v];
        out[off + 2] = p2;
    }
}

extern "C" void kernel_launch(void* const* d_in, const int* in_sizes, int n_in,
                              void* d_out, int out_size, void* d_ws, size_t ws_size,
                              hipStream_t stream) {
    const float* x = (const float*)d_in[0];   // [4,512,128] -> [2048,128]
    const float* y = (const float*)d_in[1];   // [512,128]
    float* out = (float*)d_out;               // [2048,512,3]
    (void)in_sizes; (void)n_in; (void)out_size; (void)d_ws; (void)ws_size;

    dim3 grid(kM / kBT, kR / kBT);   // (16, 64)
    dim3 block(128);                 // 4 wave32
    distance_wmma_kernel<<<grid, block, 0, stream>>>(x, y, out);
}